// GraphMaxPooling_31482110280182
// MI455X (gfx1250) — compile-verified
//
#include <hip/hip_runtime.h>

// out[i][j] = in[2i+1][2j+1], in: 8192x8192 f32, out: 4096x4096 f32.
// Pure streaming decimation: 128 MB read (odd rows only) + 64 MB write,
// roofline ~8.2 us at 23.3 TB/s. No matrix math -> no WMMA; TDM cannot
// express stride-2 along dim0, so wide coalesced global ops with NT hints
// are the optimal MI455X path.

#define N_IN   8192
#define N_OUT  4096
#define ROWS_PER_THREAD 4

typedef float v4f __attribute__((ext_vector_type(4)));
typedef float v2f __attribute__((ext_vector_type(2)));

__global__ __launch_bounds__(256) void decimate2_kernel(const float* __restrict__ in,
                                                        float* __restrict__ out) {
    // tx in [0, 2048): each lane produces 2 consecutive output columns.
    const int tx   = blockIdx.x * blockDim.x + threadIdx.x;
    const int row0 = blockIdx.y * ROWS_PER_THREAD;

    const long in_col  = 4L * tx;   // covers input cols 4tx .. 4tx+3
    const long out_col = 2L * tx;   // covers output cols 2tx, 2tx+1

#pragma unroll
    for (int r = 0; r < ROWS_PER_THREAD; ++r) {
        const long orow = row0 + r;
        const long irow = 2 * orow + 1;

        // Coalesced 16B/lane load of the odd input row (read-once -> NT).
        const v4f* src = (const v4f*)(in + irow * N_IN + in_col);
        v4f v = __builtin_nontemporal_load(src);

        // Keep odd columns: cols 4tx+1 (v.y) and 4tx+3 (v.w).
        v2f o;
        o.x = v.y;
        o.y = v.w;

        // Coalesced 8B/lane store of the output row (write-once -> NT).
        v2f* dst = (v2f*)(out + orow * N_OUT + out_col);
        __builtin_nontemporal_store(o, dst);
    }
}

extern "C" void kernel_launch(void* const* d_in, const int* in_sizes, int n_in,
                              void* d_out, int out_size, void* d_ws, size_t ws_size,
                              hipStream_t stream) {
    const float* support = (const float*)d_in[0];
    float* out = (float*)d_out;

    // 2048 lanes per output row pair-of-columns -> grid.x = 2048/256 = 8.
    // 4096 output rows / 4 rows-per-thread -> grid.y = 1024.
    dim3 block(256, 1, 1);
    dim3 grid(N_OUT / 2 / 256, N_OUT / ROWS_PER_THREAD, 1);
    decimate2_kernel<<<grid, block, 0, stream>>>(support, out);
}